// convAttention_3530463117706
// MI455X (gfx1250) — compile-verified
//
#include <hip/hip_runtime.h>

// ---------------------------------------------------------------------------
// convAttention on MI455X (gfx1250)
//
// The reference's einsum 'bhkxy,bhdxy->bhdxy' contracts the softmax over its
// own normalization axis k, so attention contributes exactly 1.0 and the
// output reduces to  y = W_proj @ (W_v @ x + b_v) + b_proj  — a single folded
// 192x192 1x1 conv over 131072 pixels (9.7 GFLOP, 201 MB => memory-bound at
// 23.3 TB/s, ~8.6us). We implement it as a streaming fp32 WMMA GEMM.
// ---------------------------------------------------------------------------

#define DIM   192
#define HW    16384        // 128*128
#define BATCH 8

typedef __attribute__((ext_vector_type(2))) float v2f;
typedef __attribute__((ext_vector_type(8))) float v8f;

// Fold the two 1x1 convs: W_eff[o][i] = sum_c W_proj[o][c] * W_qkv[384+c][i]
//                         b_eff[o]    = sum_c W_proj[o][c] * b_qkv[384+c] + b_proj[o]
__global__ __launch_bounds__(DIM) void fold_weights(
    const float* __restrict__ w_qkv,   // [576][192]
    const float* __restrict__ b_qkv,   // [576]
    const float* __restrict__ w_proj,  // [192][192]
    const float* __restrict__ b_proj,  // [192]
    float* __restrict__ w_eff,         // [192][192]
    float* __restrict__ b_eff)         // [192]
{
    const int o = blockIdx.x;          // output channel (row of W_eff)
    const int i = threadIdx.x;         // input channel  (col of W_eff)
    const float* wv = w_qkv + 384 * DIM;   // rows 384..575 produce v
    float s = 0.f;
    for (int c = 0; c < DIM; ++c)
        s = fmaf(w_proj[o * DIM + c], wv[c * DIM + i], s);
    w_eff[o * DIM + i] = s;
    if (i == 0) {
        const float* bv = b_qkv + 384;
        float t = b_proj[o];
        for (int c = 0; c < DIM; ++c)
            t = fmaf(w_proj[o * DIM + c], bv[c], t);
        b_eff[o] = t;
    }
}

// Per batch b:  Out[192, 16384] = W_eff[192,192] x X[192, 16384]  (+ b_eff)
// Block = 384 threads = 12 wave32s. Wave w computes the 16x16 output tile
// (M = w*16 .. w*16+15, N = blockIdx.x*16 .. +15) with 48 chained
// v_wmma_f32_16x16x4_f32 steps over K = 0..191.
//
// ISA 7.12.2 fp32 fragment layout (wave32):
//   A (16x4, 2 VGPRs): lane<16 -> M=lane,    K0_of_pair = 0
//                      lane>=16 -> M=lane-16, K0_of_pair = 2
//                      VGPR0 = A[M][k0+ko], VGPR1 = A[M][k0+ko+1]
//   B (4x16, 2 VGPRs): same K split across lane halves, N = lane&15
//   C/D (16x16, 8 VGPRs): VGPR r, lane<16 -> (M=r, N=lane)
//                                  lane>=16 -> (M=r+8, N=lane-16)
__global__ __launch_bounds__(384) void fused_gemm(
    const float* __restrict__ x,      // [B][192][HW]
    const float* __restrict__ w_eff,  // [192][192]
    const float* __restrict__ b_eff,  // [192]
    float* __restrict__ out)          // [B][192][HW]
{
    const int lane = threadIdx.x & 31;
    const int wave = threadIdx.x >> 5;          // 0..11 -> M tile
    const int b    = blockIdx.y;                // batch
    const int n0   = blockIdx.x * 16;           // pixel tile
    const int m0   = wave * 16;                 // output-channel tile

    const int half = lane >> 4;                 // 0 | 1 (lane half)
    const int l15  = lane & 15;
    const int ko   = half * 2;                  // K offset within group of 4

    const float* xb = x   + (size_t)b * DIM * HW;
    float*       ob = out + (size_t)b * DIM * HW;

    const float* arow = w_eff + (size_t)(m0 + l15) * DIM;  // A row this lane feeds
    const float* xcol = xb + n0 + l15;                     // X column this lane feeds

    v8f acc = {};                                // C starts at 0; bias added at store

    #pragma unroll
    for (int k0 = 0; k0 < DIM; k0 += 4) {
        v2f af;
        af.x = arow[k0 + ko];                    // contiguous pair -> b64 load
        af.y = arow[k0 + ko + 1];
        v2f bf;
        bf.x = xcol[(size_t)(k0 + ko) * HW];     // 64B-coalesced per lane half
        bf.y = xcol[(size_t)(k0 + ko + 1) * HW];
        acc = __builtin_amdgcn_wmma_f32_16x16x4_f32(
            /*neg_a=*/false, af, /*neg_b=*/false, bf,
            /*c_mod=*/(short)0, acc, /*reuse_a=*/false, /*reuse_b=*/false);
    }

    const int mbase = m0 + half * 8;
    #pragma unroll
    for (int r = 0; r < 8; ++r) {
        const int M = mbase + r;
        ob[(size_t)M * HW + n0 + l15] = acc[r] + b_eff[M];
    }
}

extern "C" void kernel_launch(void* const* d_in, const int* in_sizes, int n_in,
                              void* d_out, int out_size, void* d_ws, size_t ws_size,
                              hipStream_t stream) {
    const float* x      = (const float*)d_in[0];
    const float* w_qkv  = (const float*)d_in[1];
    const float* b_qkv  = (const float*)d_in[2];
    // d_in[3] (w_attn) and d_in[4] (b_attn) are mathematically dead:
    // softmax sums to 1 over the contracted einsum axis.
    const float* w_proj = (const float*)d_in[5];
    const float* b_proj = (const float*)d_in[6];
    float* out = (float*)d_out;

    float* w_eff = (float*)d_ws;                 // 192*192 floats
    float* b_eff = w_eff + DIM * DIM;            // 192 floats

    fold_weights<<<dim3(DIM), dim3(DIM), 0, stream>>>(
        w_qkv, b_qkv, w_proj, b_proj, w_eff, b_eff);

    fused_gemm<<<dim3(HW / 16, BATCH), dim3(384), 0, stream>>>(
        x, w_eff, b_eff, out);
}